// DWT_3D_84688165143381
// MI455X (gfx1250) — compile-verified
//
#include <hip/hip_runtime.h>

typedef __attribute__((ext_vector_type(2))) float v2f;
typedef __attribute__((ext_vector_type(8))) float v8f;

// 3D Haar DWT, (N,C,D,H,W) = (2,8,128,128,128) fp32 -> 8 sub-bands (2,8,64,64,64).
// H-pass: fp32 WMMA vs constant banded 16x16 Haar block. W-pass: lane butterfly.
// D-pass: slice-pair butterfly. Memory-bound: ~268 MiB @ 23.3 TB/s => ~11.5 us floor.
__global__ __launch_bounds__(256) void dwt3d_haar_wmma_kernel(
    const float* __restrict__ x, float* __restrict__ out)
{
    constexpr int   Wd = 128;                 // W stride of a row
    constexpr int   HW = 128 * 128;           // one D-slice
    constexpr float c  = 0.70710678118654752440f;

    const int lane = threadIdx.x & 31;
    const int wave = threadIdx.x >> 5;
    const int id   = blockIdx.x * 8 + wave;   // 65536 waves total

    const int wT = id & 7;                    // W tile   (8 tiles of 16)
    const int hT = (id >> 3) & 7;             // H tile   (8 tiles of 16)
    const int e  = (id >> 6) & 63;            // output depth index (D/2)
    const int nc = id >> 12;                  // fused N*C in [0,16)

    const int n16 = lane & 15;                // A: row m ; B: column n
    const int hi  = lane >> 4;                // K-half selector (0 or 1)

    // ---- Constant A operand: 16x16 Haar block T, chunked along K into 4x (16x4).
    // A layout (32-bit 16x4): lanes 0-15 m=0..15 K={4k,4k+1}; lanes 16-31 K={4k+2,4k+3}.
    // T rows 0-7: low  (+c at cols 2m,2m+1); rows 8-15: high (+c at 2(m-8), -c at 2(m-8)+1).
    v2f A[4];
#pragma unroll
    for (int k = 0; k < 4; ++k) {
        const int jh = 2 * k + hi;            // column-pair index j>>1 for {.x,.y}
        float ax = 0.0f, ay = 0.0f;
        if (n16 < 8) {
            if (jh == n16)      { ax = c;  ay = c;  }
        } else {
            if (jh == n16 - 8)  { ax = c;  ay = -c; }
        }
        A[k].x = ax; A[k].y = ay;
    }

    // ---- Load the two input tiles (slices 2e, 2e+1) directly in B-operand layout.
    // B (32-bit 4x16): vgpr v, lane l -> K = 4k + 2*(l>=16) + v, N = l&15.
    const long long baseIn = (long long)nc * (128LL * HW)
                           + (long long)(2 * e) * HW
                           + (long long)(hT * 16) * Wd
                           + (long long)(wT * 16) + n16;
    const float* p0 = x + baseIn;             // slice 2e
    const float* p1 = p0 + HW;                // slice 2e+1

    v2f B0[4], B1[4];
#pragma unroll
    for (int k = 0; k < 4; ++k) {
        const int r0 = 4 * k + 2 * hi;        // tile row for .x; .y is row r0+1
        B0[k].x = p0[(long long)r0 * Wd];
        B0[k].y = p0[(long long)(r0 + 1) * Wd];
        B1[k].x = p1[(long long)r0 * Wd];
        B1[k].y = p1[(long long)(r0 + 1) * Wd];
    }

    // ---- H transform: V = T * X  (16x16x16 fp32 via 4 chained 16x16x4 WMMAs).
    v8f V0 = {};
    v8f V1 = {};
#pragma unroll
    for (int k = 0; k < 4; ++k)
        V0 = __builtin_amdgcn_wmma_f32_16x16x4_f32(false, A[k], false, B0[k],
                                                   (short)0, V0, false, false);
#pragma unroll
    for (int k = 0; k < 4; ++k)
        V1 = __builtin_amdgcn_wmma_f32_16x16x4_f32(false, A[k], false, B1[k],
                                                   (short)0, V1, false, false);

    // ---- D transform (slice-pair butterfly): P = low-D, M = high-D.
    v8f P, M;
#pragma unroll
    for (int r = 0; r < 8; ++r) { P[r] = V0[r] + V1[r]; M[r] = V0[r] - V1[r]; }

    // ---- W transform (adjacent-lane butterfly) + scatter to sub-bands.
    // C layout: vgpr r, lane l -> tile row m' = r + 8*hi, tile col n = l&15.
    // Output coords: p = hT*8 + r (H-band = hi), q = wT*8 + n/2 (W-band = n&1).
    const int   wband      = lane & 1;
    const int   q          = wT * 8 + (n16 >> 1);
    const long long bandSz = 16LL * 64 * 64 * 64;            // elems per sub-band
    const long long inner  = ((long long)nc * 64 + e) * 4096 // nc,e part
                           + (long long)q;
    const int bLo = hi * 2 + wband;                          // D-low sub-band
#pragma unroll
    for (int r = 0; r < 8; ++r) {
        const long long rowOff = inner + (long long)(hT * 8 + r) * 64;

        const float vP = P[r];
        const float oP = __shfl_xor(vP, 1, 32);
        const float tP = wband ? (oP - vP) : (vP + oP);      // high: x[2i]-x[2i+1]
        __builtin_nontemporal_store(0.5f * tP, out + (long long)bLo * bandSz + rowOff);

        const float vM = M[r];
        const float oM = __shfl_xor(vM, 1, 32);
        const float tM = wband ? (oM - vM) : (vM + oM);
        __builtin_nontemporal_store(0.5f * tM, out + (long long)(bLo + 4) * bandSz + rowOff);
    }
}

extern "C" void kernel_launch(void* const* d_in, const int* in_sizes, int n_in,
                              void* d_out, int out_size, void* d_ws, size_t ws_size,
                              hipStream_t stream) {
    const float* x   = (const float*)d_in[0];   // data (2,8,128,128,128) fp32
    float*       out = (float*)d_out;           // 8 bands x (2,8,64,64,64) fp32, concat
    // 65536 waves, 8 waves (256 threads) per block.
    dwt3d_haar_wmma_kernel<<<8192, 256, 0, stream>>>(x, out);
}